// Softmax_292057776662
// MI455X (gfx1250) — compile-verified
//
#include <hip/hip_runtime.h>
#include <hip/hip_bf16.h>
#include <stdint.h>

// ---------------------------------------------------------------------------
// Margin softmax + stable 4-class counting sort, fused.  MI455X (gfx1250).
// Bandwidth-bound (~0.37 GB total traffic -> ~16us floor @ 23.3 TB/s).
// CDNA5 paths used: global_load_async_to_lds_b128 (ASYNCcnt double-buffer),
// s_wait_asynccnt, wave32 ballots, non-temporal stores, global_prefetch.
// ---------------------------------------------------------------------------

typedef float v4f __attribute__((ext_vector_type(4)));

#define BLOCK 256
#define CHUNK 4096
#define ITERS (CHUNK / BLOCK)   // 16
#define WAVES (BLOCK / 32)      // 8 (wave32)

#define MARG_M   4.0f
#define MARG_N   0.5f
#define MARG_EPS 1e-5f

// Async copy of one 16B row into this lane's LDS slot.  LDS byte address is
// the low 32 bits of the generic pointer (HW flat->LDS mapping).
__device__ __forceinline__ void async_row_to_lds(const float* g, v4f* l) {
    unsigned loff = (unsigned)(uintptr_t)l;
    unsigned long long ga = (unsigned long long)(uintptr_t)g;
    asm volatile("global_load_async_to_lds_b128 %0, %1, off"
                 :: "v"(loff), "v"(ga) : "memory");
}

// ------------- pass 3: stable scatter + margin + log-softmax ---------------
// (defined first so the disasm snippet shows the async-copy double buffer)
__global__ __launch_bounds__(BLOCK) void k_scatter(
        const float* __restrict__ data, const int* __restrict__ label,
        const int* __restrict__ offsets, int numBlocks, int N,
        float* __restrict__ out, float* __restrict__ lossPartial)
{
    __shared__ v4f   s_row[2][BLOCK];     // async double buffer (8 KB)
    __shared__ int   s_wcnt[WAVES][4];
    __shared__ float s_loss[WAVES];
    const int tid = threadIdx.x, lane = tid & 31, wave = tid >> 5;
    const int b = blockIdx.x;
    const int base = b * CHUNK;
    const int off0 = offsets[0 * numBlocks + b];
    const int off1 = offsets[1 * numBlocks + b];
    const int off2 = offsets[2 * numBlocks + b];
    const int off3 = offsets[3 * numBlocks + b];
    int run0 = 0, run1 = 0, run2 = 0, run3 = 0;
    float lossAcc = 0.f;
    float* __restrict__ outLabel = out + (size_t)4 * N;

    {   // prologue: stage iteration 0's rows via the async copy engine
        int idx = base + tid;
        if (idx < N) async_row_to_lds(data + (size_t)idx * 4, &s_row[0][tid]);
    }
#pragma unroll 1
    for (int j = 0; j < ITERS; ++j) {
        int idx = base + j * BLOCK + tid;
        bool valid = idx < N;
        int c = valid ? label[idx] : -1;
        unsigned m0 = (unsigned)__ballot(c == 0);
        unsigned m1 = (unsigned)__ballot(c == 1);
        unsigned m2 = (unsigned)__ballot(c == 2);
        unsigned m3 = (unsigned)__ballot(c == 3);
        if (lane == 0) {
            s_wcnt[wave][0] = __popc(m0); s_wcnt[wave][1] = __popc(m1);
            s_wcnt[wave][2] = __popc(m2); s_wcnt[wave][3] = __popc(m3);
        }
        // issue next tile's async load, then drain down to the current tile
        if (j + 1 < ITERS) {
            int nidx = base + (j + 1) * BLOCK + tid;
            if (nidx < N)
                async_row_to_lds(data + (size_t)nidx * 4, &s_row[(j + 1) & 1][tid]);
            asm volatile("s_wait_asynccnt 1" ::: "memory");
        } else {
            asm volatile("s_wait_asynccnt 0" ::: "memory");
        }
        __syncthreads();                           // s_wcnt visible
        int t0 = 0, t1 = 0, t2 = 0, t3 = 0, wp = 0;
#pragma unroll
        for (int w = 0; w < WAVES; ++w) {
            int a0 = s_wcnt[w][0], a1 = s_wcnt[w][1], a2 = s_wcnt[w][2], a3 = s_wcnt[w][3];
            if (w < wave) wp += (c == 0 ? a0 : c == 1 ? a1 : c == 2 ? a2 : c == 3 ? a3 : 0);
            t0 += a0; t1 += a1; t2 += a2; t3 += a3;
        }
        v4f r = s_row[j & 1][tid];                 // loaded by this lane's async op
        if (valid) {
            unsigned below = (1u << lane) - 1u;
            unsigned mc = (c == 0) ? m0 : (c == 1) ? m1 : (c == 2) ? m2 : m3;
            int lrank = __popc(mc & below) + wp;   // stable rank in block
            int dst = ((c == 0) ? off0 + run0 : (c == 1) ? off1 + run1
                     : (c == 2) ? off2 + run2 : off3 + run3) + lrank;
            // margin-adjust the true-class logit
            float v = (c == 0) ? r.x : (c == 1) ? r.y : (c == 2) ? r.z : r.w;
            float adj = (v > 0.f) ? (v / (MARG_M + MARG_EPS) - MARG_N)
                                  : (v * (MARG_M + MARG_EPS) - MARG_N);
            if (c == 0) r.x = adj; else if (c == 1) r.y = adj;
            else if (c == 2) r.z = adj; else r.w = adj;
            // 4-wide log-softmax -> NLL of true class
            float mx = fmaxf(fmaxf(r.x, r.y), fmaxf(r.z, r.w));
            float s = __expf(r.x - mx) + __expf(r.y - mx) +
                      __expf(r.z - mx) + __expf(r.w - mx);
            float lse = mx + __logf(s);
            lossAcc += (lse - adj);
            // streaming outputs: never re-read -> non-temporal
            __builtin_nontemporal_store(r, (v4f*)out + dst);
            __builtin_nontemporal_store((float)c, outLabel + dst);
        }
        run0 += t0; run1 += t1; run2 += t2; run3 += t3;
        __syncthreads();                           // protect s_wcnt reuse
    }
#pragma unroll
    for (int d = 16; d > 0; d >>= 1) lossAcc += __shfl_down(lossAcc, d, 32);
    if (lane == 0) s_loss[wave] = lossAcc;
    __syncthreads();
    if (tid == 0) {
        float t = 0.f;
#pragma unroll
        for (int w = 0; w < WAVES; ++w) t += s_loss[w];
        lossPartial[b] = t;
    }
}

// ------------------------------ pass 1: histogram ---------------------------
// One int4 (b128) label load per thread per iteration; one 64B-line prefetch
// of the matching 4 rows of data to pre-warm the 192MB L2 for the scatter.
__global__ __launch_bounds__(BLOCK) void k_hist(
        const int* __restrict__ label, const float* __restrict__ data,
        int N, int numBlocks, int* __restrict__ counts)
{
    const int tid = threadIdx.x;
    const int lane = tid & 31, wave = tid >> 5;
    const int base4 = blockIdx.x * (CHUNK / 4);     // in int4 units
    const int4* __restrict__ lab4 = (const int4*)label;
    int c0 = 0, c1 = 0, c2 = 0, c3 = 0;
#pragma unroll
    for (int j = 0; j < ITERS / 4; ++j) {           // 4 iterations
        int vidx = base4 + j * BLOCK + tid;
        int idx0 = vidx * 4;
        if (idx0 + 3 < N) {
            int4 L = lab4[vidx];
            c0 += (L.x == 0) + (L.y == 0) + (L.z == 0) + (L.w == 0);
            c1 += (L.x == 1) + (L.y == 1) + (L.z == 1) + (L.w == 1);
            c2 += (L.x == 2) + (L.y == 2) + (L.z == 2) + (L.w == 2);
            c3 += (L.x == 3) + (L.y == 3) + (L.z == 3) + (L.w == 3);
            __builtin_prefetch(data + (size_t)idx0 * 4, 0, 1);  // 4 rows = 64B
        } else {
#pragma unroll 1
            for (int k = 0; k < 4; ++k) {
                int idx = idx0 + k;
                if (idx < N) {
                    int c = label[idx];
                    c0 += (c == 0); c1 += (c == 1); c2 += (c == 2); c3 += (c == 3);
                }
            }
        }
    }
#pragma unroll
    for (int d = 16; d > 0; d >>= 1) {
        c0 += __shfl_down(c0, d, 32);
        c1 += __shfl_down(c1, d, 32);
        c2 += __shfl_down(c2, d, 32);
        c3 += __shfl_down(c3, d, 32);
    }
    __shared__ int s[WAVES][4];
    if (lane == 0) { s[wave][0] = c0; s[wave][1] = c1; s[wave][2] = c2; s[wave][3] = c3; }
    __syncthreads();
    if (tid < 4) {
        int t = 0;
#pragma unroll
        for (int w = 0; w < WAVES; ++w) t += s[w][tid];
        counts[tid * numBlocks + blockIdx.x] = t;   // class-major
    }
}

// --------------------- pass 2: exclusive scan (one block) -------------------
// Full exclusive scan over class-major counts == classBase + per-class prefix.
__global__ __launch_bounds__(1024) void k_scan(
        const int* __restrict__ counts, int M, int* __restrict__ offsets)
{
    const int tid = threadIdx.x;
    const int lane = tid & 31, wave = tid >> 5;
    const int G = (M + 1023) >> 10;
    const int start = tid * G;
    int sum = 0;
    for (int i = 0; i < G; ++i) {
        int p = start + i;
        if (p < M) sum += counts[p];
    }
    int v = sum;
#pragma unroll
    for (int d = 1; d < 32; d <<= 1) {
        int n = __shfl_up(v, d, 32);
        if (lane >= d) v += n;
    }
    __shared__ int wsum[32];
    if (lane == 31) wsum[wave] = v;
    __syncthreads();
    if (wave == 0) {
        int w = wsum[lane];
#pragma unroll
        for (int d = 1; d < 32; d <<= 1) {
            int n = __shfl_up(w, d, 32);
            if (lane >= d) w += n;
        }
        wsum[lane] = w;
    }
    __syncthreads();
    int run = ((wave == 0) ? 0 : wsum[wave - 1]) + (v - sum);   // exclusive base
    for (int i = 0; i < G; ++i) {
        int p = start + i;
        if (p < M) { offsets[p] = run; run += counts[p]; }
    }
}

// --------------------------- pass 4: final reduce ---------------------------
__global__ __launch_bounds__(256) void k_final(
        const float* __restrict__ lossPartial, int numBlocks, int N,
        float* __restrict__ lossOut)
{
    __shared__ float sm[256];
    const int tid = threadIdx.x;
    float a = 0.f;
    for (int i = tid; i < numBlocks; i += 256) a += lossPartial[i];
    sm[tid] = a;
    __syncthreads();
#pragma unroll
    for (int s = 128; s > 0; s >>= 1) {
        if (tid < s) sm[tid] += sm[tid + s];
        __syncthreads();
    }
    if (tid == 0) lossOut[0] = sm[0] / (float)N;
}

// ---------------------------------------------------------------------------
extern "C" void kernel_launch(void* const* d_in, const int* in_sizes, int n_in,
                              void* d_out, int out_size, void* d_ws, size_t ws_size,
                              hipStream_t stream) {
    const float* data  = (const float*)d_in[0];   // [N,4] f32
    const int*   label = (const int*)d_in[1];     // [N]   i32
    const int N = in_sizes[1];
    const int numBlocks = (N + CHUNK - 1) / CHUNK;
    const int M = 4 * numBlocks;

    int*   counts      = (int*)d_ws;              // [4][numBlocks] class-major
    int*   offsets     = counts + M;              // [4][numBlocks]
    float* lossPartial = (float*)(offsets + M);   // [numBlocks]

    float* out     = (float*)d_out;               // data(4N) | label(N) | loss(1)
    float* lossOut = out + (size_t)5 * N;

    k_hist   <<<numBlocks, BLOCK, 0, stream>>>(label, data, N, numBlocks, counts);
    k_scan   <<<1,         1024,  0, stream>>>(counts, M, offsets);
    k_scatter<<<numBlocks, BLOCK, 0, stream>>>(data, label, offsets, numBlocks, N,
                                               out, lossPartial);
    k_final  <<<1,         256,   0, stream>>>(lossPartial, numBlocks, N, lossOut);
}